// PlaceEngine_49065706390061
// MI455X (gfx1250) — compile-verified
//
#include <hip/hip_runtime.h>
#include <float.h>

// CDNA5 / gfx1250 — wave32, WMMA f32 16x16x4 for pairwise squared distances.

typedef float v2f __attribute__((ext_vector_type(2)));
typedef float v8f __attribute__((ext_vector_type(8)));

#define NPTS        8192
#define TILES_DIM   512          // 8192 / 16
#define WAVES_PB    8            // 256 threads = 8 wave32
#define BLOCKS      2048         // 2048*8 = 16384 waves
#define COLGRPS     32           // waves sharing one row-tile
#define TILES_PW    16           // column tiles per wave (512/32)

__global__ __launch_bounds__(256)
void stress_tiles_kernel(const float* __restrict__ pos,
                         const float* __restrict__ dist,
                         float* __restrict__ partial) {
    const int lane = threadIdx.x & 31;
    const int wid  = blockIdx.x * WAVES_PB + (threadIdx.x >> 5);   // 0..16383
    const int row_tile = wid >> 5;          // 0..511
    const int colgrp   = wid & (COLGRPS - 1);

    const int  m  = lane & 15;
    const bool hi = lane >= 16;

    // ---- A fragment: rows of this tile, homogeneous coords [x, y, n2, 1] ----
    // 16x4 f32 A layout: lanes 0-15 hold K=0 (V0) and K=1 (V1) for M=lane;
    //                    lanes 16-31 hold K=2 (V0) and K=3 (V1) for M=lane-16.
    const float2 pr = ((const float2*)pos)[row_tile * 16 + m];
    const float n2r = fmaf(pr.x, pr.x, pr.y * pr.y);
    v2f A;
    A.x = hi ? n2r  : pr.x;   // K=0 -> x_m   | K=2 -> ||p_m||^2
    A.y = hi ? 1.0f : pr.y;   // K=1 -> y_m   | K=3 -> 1

    float acc0 = 0.0f, acc1 = 0.0f;   // two chains -> packed v_pk_add_f32

    for (int ct = 0; ct < TILES_PW; ++ct) {
        const int col_tile = colgrp * TILES_PW + ct;

        // ---- B fragment: cols of this tile, [-2x, -2y, 1, n2] ----
        const float2 pc = ((const float2*)pos)[col_tile * 16 + m];
        const float n2c = fmaf(pc.x, pc.x, pc.y * pc.y);
        v2f B;
        B.x = hi ? 1.0f : -2.0f * pc.x;   // K=0 row -> -2x_n | K=2 row -> 1
        B.y = hi ? n2c  : -2.0f * pc.y;   // K=1 row -> -2y_n | K=3 row -> n2_n

        // sq[m][n] = n2_m + n2_n - 2 p_m.p_n = ||p_m - p_n||^2
        v8f sq = {};
        sq = __builtin_amdgcn_wmma_f32_16x16x4_f32(
                 false, A, false, B, (short)0, sq, false, false);

        // ---- dist tile, WMMA C/D layout: VGPR r -> row (r + 8*hi), col = m ----
        const float* dbase = dist
            + (size_t)(row_tile * 16 + (hi ? 8 : 0)) * NPTS
            + (size_t)(col_tile * 16 + m);

#pragma unroll
        for (int r = 0; r < 8; ++r) {
            const float d = __builtin_nontemporal_load(dbase + (size_t)r * NPTS);
            // clamp(sq, 0, +MAX) in ONE v_med3_f32 (no canonicalize pass):
            const float s    = __builtin_amdgcn_fmed3f(sq[r], 0.0f, FLT_MAX);
            const float pred = __builtin_amdgcn_sqrtf(s);
            const float rcpd = __builtin_amdgcn_rcpf(d);
            const float e    = fmaf(pred, rcpd, -1.0f);
            const float c    = (d != 0.0f) ? e * e : 0.0f;  // dist==0 -> contributes 0
            if (r & 1) acc1 += c; else acc0 += c;
        }
    }

    // ---- deterministic block reduction ----
    __shared__ float red[256];
    red[threadIdx.x] = acc0 + acc1;
    __syncthreads();
#pragma unroll
    for (int s = 128; s > 0; s >>= 1) {
        if (threadIdx.x < s) red[threadIdx.x] += red[threadIdx.x + s];
        __syncthreads();
    }
    if (threadIdx.x == 0) partial[blockIdx.x] = red[0];
}

__global__ __launch_bounds__(256)
void final_reduce_kernel(const float* __restrict__ partial, float* __restrict__ out) {
    __shared__ float red[256];
    float a = 0.0f;
    for (int i = threadIdx.x; i < BLOCKS; i += 256) a += partial[i];
    red[threadIdx.x] = a;
    __syncthreads();
#pragma unroll
    for (int s = 128; s > 0; s >>= 1) {
        if (threadIdx.x < s) red[threadIdx.x] += red[threadIdx.x + s];
        __syncthreads();
    }
    if (threadIdx.x == 0) out[0] = red[0];
}

extern "C" void kernel_launch(void* const* d_in, const int* in_sizes, int n_in,
                              void* d_out, int out_size, void* d_ws, size_t ws_size,
                              hipStream_t stream) {
    (void)in_sizes; (void)n_in; (void)out_size; (void)ws_size;
    const float* pos  = (const float*)d_in[0];   // [8192, 2] f32
    const float* dist = (const float*)d_in[1];   // [8192, 8192] f32
    float* partial = (float*)d_ws;               // 2048 floats of scratch
    float* out     = (float*)d_out;              // scalar f32

    stress_tiles_kernel<<<BLOCKS, 256, 0, stream>>>(pos, dist, partial);
    final_reduce_kernel<<<1, 256, 0, stream>>>(partial, out);
}